// SoftReRank_64201171141092
// MI455X (gfx1250) — compile-verified
//
#include <hip/hip_runtime.h>
#include <stdint.h>

// Problem geometry (from reference): x is [4096, 16384] fp32,
// output is [4096, 16] fp32 = concat(8 smallest asc, 8 largest asc).
constexpr int ROW    = 16384;   // floats per row
constexpr int TPB    = 256;    // threads per block (8 wave32)
constexpr int CHUNK  = 4096;   // floats per LDS chunk (16 KB)
constexpr int NCHUNK = ROW / CHUNK;        // 4
constexpr int PERJ   = CHUNK / TPB / 4;    // unused sanity
constexpr int NSEG   = CHUNK / (TPB * 4);  // 4 b128 segments per thread per chunk

// Compare-exchange (ascending)
__device__ __forceinline__ void cx(float& a, float& b) {
    float lo = fminf(a, b);
    float hi = fmaxf(a, b);
    a = lo; b = hi;
}

// Sort a *bitonic* sequence of 8 ascending (bitonic merge network, 12 CX).
__device__ __forceinline__ void bitonic8_asc(float v[8]) {
    cx(v[0], v[4]); cx(v[1], v[5]); cx(v[2], v[6]); cx(v[3], v[7]);
    cx(v[0], v[2]); cx(v[1], v[3]); cx(v[4], v[6]); cx(v[5], v[7]);
    cx(v[0], v[1]); cx(v[2], v[3]); cx(v[4], v[5]); cx(v[6], v[7]);
}

// Update sorted-ascending min8 (m[7] = largest kept) and max8 (M[0] = smallest kept).
__device__ __forceinline__ void upd(float v, float m[8], float M[8]) {
    if (v < m[7]) {             // insert into min list, drop old m[7]
        m[7] = v;
        #pragma unroll
        for (int i = 6; i >= 0; --i) cx(m[i], m[i + 1]);
    }
    if (v > M[0]) {             // insert into max list, drop old M[0]
        M[0] = v;
        #pragma unroll
        for (int i = 0; i < 7; ++i) cx(M[i], M[i + 1]);
    }
}

__global__ __launch_bounds__(TPB) void softrerank_kernel(const float* __restrict__ x,
                                                         float* __restrict__ out) {
    __shared__ float smem[2 * CHUNK];   // 32 KB: double-buffered stream chunks
                                        // (reduction region aliased in later)
    const int tid = threadIdx.x;
    const int row = blockIdx.x;
    const float* rowp = x + (size_t)row * ROW;

    float mn[8], mx[8];
    #pragma unroll
    for (int i = 0; i < 8; ++i) { mn[i] =  __builtin_inff(); mx[i] = -__builtin_inff(); }

    // Issue async B128 copies for chunk c into LDS buffer `buf`.
    // Thread t copies exactly the 4x float4 it will later read -> no barriers needed;
    // per-wave ASYNCcnt ordering is the only synchronization.
    auto issue_chunk = [&](int c, int buf) {
        const float* g = rowp + c * CHUNK;
        #pragma unroll
        for (int j = 0; j < NSEG; ++j) {
            const int off = j * (TPB * 4) + tid * 4;           // float index in chunk
            unsigned lds = (unsigned)(uintptr_t)(&smem[buf * CHUNK + off]);
            unsigned long long ga = (unsigned long long)(uintptr_t)(g + off);
            asm volatile("global_load_async_to_lds_b128 %0, %1, off"
                         :
                         : "v"(lds), "v"(ga)
                         : "memory");
        }
    };

    issue_chunk(0, 0);
    for (int c = 0; c < NCHUNK; ++c) {
        const int buf = c & 1;
        if (c + 1 < NCHUNK) {
            issue_chunk(c + 1, (c + 1) & 1);
            // 4 outstanding asyncs belong to chunk c+1; async loads complete
            // in order, so ASYNCcnt<=4 means chunk c's data is in LDS.
            asm volatile("s_wait_asynccnt 0x4" ::: "memory");
        } else {
            asm volatile("s_wait_asynccnt 0x0" ::: "memory");
        }
        // Consume chunk c from LDS (the exact bytes this thread copied).
        #pragma unroll
        for (int j = 0; j < NSEG; ++j) {
            const int off = j * (TPB * 4) + tid * 4;
            const float4 v = *reinterpret_cast<const float4*>(&smem[buf * CHUNK + off]);
            upd(v.x, mn, mx); upd(v.y, mn, mx); upd(v.z, mn, mx); upd(v.w, mn, mx);
        }
    }

    // ---- Cross-thread reduction: 256 sorted 8-lists -> one, via bitonic split ----
    float* red = smem;                  // alias (16 KB needed, 32 KB available)
    __syncthreads();                    // everyone done reading stream buffers
    #pragma unroll
    for (int i = 0; i < 8; ++i) { red[tid * 16 + i] = mn[i]; red[tid * 16 + 8 + i] = mx[i]; }
    __syncthreads();

    for (int s = TPB / 2; s >= 1; s >>= 1) {
        if (tid < s) {
            const float* pb = &red[(tid + s) * 16];
            float a[8], b[8];
            // 8 smallest of (mn ∪ partner-mn): elementwise min against reversed
            // partner gives a bitonic sequence containing exactly the 8 smallest.
            #pragma unroll
            for (int i = 0; i < 8; ++i) a[i] = fminf(mn[i], pb[7 - i]);
            // 8 largest of (mx ∪ partner-mx): elementwise max against reversed partner.
            #pragma unroll
            for (int i = 0; i < 8; ++i) b[i] = fmaxf(mx[i], pb[8 + (7 - i)]);
            bitonic8_asc(a);
            bitonic8_asc(b);
            #pragma unroll
            for (int i = 0; i < 8; ++i) { mn[i] = a[i]; mx[i] = b[i]; }
            #pragma unroll
            for (int i = 0; i < 8; ++i) { red[tid * 16 + i] = mn[i]; red[tid * 16 + 8 + i] = mx[i]; }
        }
        __syncthreads();
    }

    if (tid == 0) {
        float* o = out + (size_t)row * 16;
        #pragma unroll
        for (int i = 0; i < 8; ++i) { o[i] = mn[i]; o[8 + i] = mx[i]; }
    }
}

extern "C" void kernel_launch(void* const* d_in, const int* in_sizes, int n_in,
                              void* d_out, int out_size, void* d_ws, size_t ws_size,
                              hipStream_t stream) {
    (void)n_in; (void)d_ws; (void)ws_size; (void)out_size;
    const float* x = (const float*)d_in[0];
    float* out = (float*)d_out;
    const int rows = in_sizes[0] / ROW;   // 4096
    softrerank_kernel<<<rows, TPB, 0, stream>>>(x, out);
}